// GeneralAttention_6167573037358
// MI455X (gfx1250) — compile-verified
//
#include <hip/hip_runtime.h>
#include <hip/hip_bf16.h>

typedef float v2f __attribute__((ext_vector_type(2)));
typedef float v8f __attribute__((ext_vector_type(8)));

#define B_       8
#define LQ       512
#define LK       4096
#define DQK      64
#define DV_      64
#define KEEP     32
#define ROWS     16
#define WAVES    16
#define NTHREADS 512
#define NEG_INF (-3.402823466e+38f)

// Dynamic LDS layout (bytes):
//   scores : ROWS*LK*4   = 262144
//   meanb  : ROWS*DV*4   =   4096
//   selv   : ROWS*KEEP*4 =   2048
//   seli   : ROWS*KEEP*4 =   2048
// total 270336 < 320KB WGP LDS
static constexpr size_t LDS_BYTES =
    (size_t)(ROWS * LK + ROWS * DV_ + ROWS * KEEP) * 4 + (size_t)ROWS * KEEP * 4;

__global__ __launch_bounds__(NTHREADS)
void general_attention_kernel(const float* __restrict__ q,
                              const float* __restrict__ k,
                              const float* __restrict__ v,
                              const unsigned char* __restrict__ mask,
                              float* __restrict__ out)
{
    extern __shared__ char smem_raw[];
    float* sc    = (float*)smem_raw;             // [ROWS][LK] masked scores
    float* meanb = sc + ROWS * LK;               // [ROWS][DV] mask*V accumulator
    float* selv  = meanb + ROWS * DV_;           // [ROWS][KEEP] selected scores
    int*   seli  = (int*)(selv + ROWS * KEEP);   // [ROWS][KEEP] selected indices

    const int tid   = threadIdx.x;
    const int wave  = tid >> 5;
    const int lane  = tid & 31;
    const int laneM = lane & 15;            // M (or N/dv-col) index within half-wave
    const int koff  = (lane >> 4) << 1;     // A/B f32 frag: lanes 0-15 -> K 0,1 ; 16-31 -> K 2,3
    const int mhalf = (lane >> 4) << 3;     // C frag: lanes 16-31 hold M+8

    const int b  = blockIdx.x >> 5;         // LQ/ROWS = 32 query tiles per batch
    const int qt = blockIdx.x & 31;
    const int row_base = b * LQ + qt * ROWS;   // flat row in (B*Lq, L) space

    // ---- phase 0: zero mean accumulator ----
    for (int i = tid; i < ROWS * DV_; i += NTHREADS) meanb[i] = 0.0f;
    __syncthreads();

    // ---- phase 1: masked scores = (Q K^T) * 0.125 via f32 WMMA (exact) ----
    // 4 independent accumulators break the WMMA->WMMA RAW chain (4-way ILP on
    // the matrix pipe) instead of a 16-deep serial accumulator dependency.
    v2f aq[16];
    {
        const float* qrow = q + (size_t)(row_base + laneM) * DQK;
        #pragma unroll
        for (int c = 0; c < 16; ++c)
            aq[c] = *(const v2f*)(qrow + 4 * c + koff);
    }
    for (int kt = wave; kt < LK / 16; kt += WAVES) {
        const float* krow = k + ((size_t)b * LK + kt * 16 + laneM) * DQK;
        v2f bk[16];
        #pragma unroll
        for (int c = 0; c < 16; ++c)
            bk[c] = *(const v2f*)(krow + 4 * c + koff);

        v8f pacc[4];
        #pragma unroll
        for (int p = 0; p < 4; ++p) {
            #pragma unroll
            for (int r = 0; r < 8; ++r) pacc[p][r] = 0.0f;
        }
        #pragma unroll
        for (int c = 0; c < 4; ++c) {        // interleave: accumulator p takes chunk 4c+p
            #pragma unroll
            for (int p = 0; p < 4; ++p) {
                const int cc = 4 * c + p;
                pacc[p] = __builtin_amdgcn_wmma_f32_16x16x4_f32(false, aq[cc], false, bk[cc],
                                                                (short)0, pacc[p],
                                                                false, false);
            }
        }

        const int l0 = kt * 16 + laneM;     // this lane's N (k position)
        #pragma unroll
        for (int r = 0; r < 8; ++r) {
            const int M = r + mhalf;
            const float s = (pacc[0][r] + pacc[1][r]) + (pacc[2][r] + pacc[3][r]);
            const unsigned char mb = mask[(size_t)(row_base + M) * LK + l0];
            sc[M * LK + l0] = mb ? s * 0.125f : NEG_INF;
        }
    }

    // ---- phase 2: mean pool partials: (mask as f32) x V via f32 WMMA ----
    // 4 dv-tile accumulators are already independent -> 4-way ILP per chunk.
    {
        v8f accm[4];
        #pragma unroll
        for (int t = 0; t < 4; ++t)
            #pragma unroll
            for (int r = 0; r < 8; ++r) accm[t][r] = 0.0f;

        const int lbase = wave * (LK / WAVES);       // 256 l-values per wave
        const unsigned char* mrow = mask + (size_t)(row_base + laneM) * LK;
        for (int cl = 0; cl < (LK / WAVES) / 4; ++cl) {
            const int l0 = lbase + cl * 4 + koff;    // this lane's K pair: l0, l0+1
            v2f am;
            am.x = (float)mrow[l0];
            am.y = (float)mrow[l0 + 1];
            const float* v0 = v + ((size_t)b * LK + l0) * DV_ + laneM;
            #pragma unroll
            for (int t = 0; t < 4; ++t) {
                v2f bv;
                bv.x = v0[t * 16];                   // V[l0][t*16+laneM]
                bv.y = v0[DV_ + t * 16];             // V[l0+1][t*16+laneM]
                accm[t] = __builtin_amdgcn_wmma_f32_16x16x4_f32(false, am, false, bv,
                                                                (short)0, accm[t],
                                                                false, false);
            }
        }
        #pragma unroll
        for (int t = 0; t < 4; ++t)
            #pragma unroll
            for (int r = 0; r < 8; ++r)
                atomicAdd(&meanb[(r + mhalf) * DV_ + t * 16 + laneM], accm[t][r]);
    }
    __syncthreads();

    // ---- phase 3: wave `w` owns query row `w`: count, top-32, softmax, output ----
    {
        const int row = wave;
        const size_t rg = (size_t)(row_base + row);

        // mask count (bytes are 0/1)
        const unsigned int* mrow32 = (const unsigned int*)(mask + rg * LK);
        unsigned int cnt = 0;
        #pragma unroll 8
        for (int j = 0; j < 32; ++j) {
            unsigned int x = mrow32[lane + 32 * j] & 0x01010101u;
            cnt += (x * 0x01010101u) >> 24;
        }
        #pragma unroll
        for (int off = 16; off; off >>= 1) cnt += (unsigned)__shfl_xor((int)cnt, off, 32);
        const float count = fmaxf((float)cnt, 1.0f);

        // exact top-32: per-lane local max over owned stride-32 indices,
        // wave argmax, winner invalidates + rescans its own slice
        float* srow = sc + row * LK;
        float lmax = NEG_INF; int lidx = -1;
        #pragma unroll 8
        for (int j = 0; j < 128; ++j) {
            const int i = lane + 32 * j;
            const float s = srow[i];
            if (s > lmax) { lmax = s; lidx = i; }
        }
        for (int sel = 0; sel < KEEP; ++sel) {
            float m = lmax; int mi = lidx;
            #pragma unroll
            for (int off = 16; off; off >>= 1) {
                const float om = __shfl_xor(m, off, 32);
                const int   oi = __shfl_xor(mi, off, 32);
                if (om > m || (om == m && oi >= 0 && (mi < 0 || oi < mi))) { m = om; mi = oi; }
            }
            if (lane == 0) { selv[row * KEEP + sel] = m; seli[row * KEEP + sel] = mi; }
            if (mi >= 0 && mi == lidx) {          // unique owner lane rescans
                srow[mi] = NEG_INF;
                lmax = NEG_INF; lidx = -1;
                #pragma unroll 8
                for (int j = 0; j < 128; ++j) {
                    const int i = lane + 32 * j;
                    const float s = srow[i];
                    if (s > lmax) { lmax = s; lidx = i; }
                }
            }
        }

        // softmax over the 32 picks (all mask-true for this input regime;
        // NEG_INF picks decay to weight 0 which matches the reference)
        const float val = selv[row * KEEP + lane];
        const float m0  = selv[row * KEEP + 0];      // first pick is the max
        const float wexp = expf(val - m0);
        float ssum = wexp;
        #pragma unroll
        for (int off = 16; off; off >>= 1) ssum += __shfl_xor(ssum, off, 32);
        const float attn = wexp / ssum;

        // weighted gather of V rows + mean, lane covers dv = lane and lane+32
        float ctx0 = 0.f, ctx1 = 0.f;
        for (int j = 0; j < KEEP; ++j) {
            const float aj = __shfl(attn, j, 32);
            int idx = seli[row * KEEP + j];
            idx = idx < 0 ? 0 : idx;                 // weight is 0 in that case
            const float* vr = v + ((size_t)b * LK + idx) * DV_;
            ctx0 += aj * vr[lane];
            ctx1 += aj * vr[lane + 32];
        }
        const float inv_c = 1.0f / count;
        float* orow = out + rg * DV_;
        orow[lane]      = meanb[row * DV_ + lane]      * inv_c + ctx0;
        orow[lane + 32] = meanb[row * DV_ + lane + 32] * inv_c + ctx1;
    }
}

extern "C" void kernel_launch(void* const* d_in, const int* in_sizes, int n_in,
                              void* d_out, int out_size, void* d_ws, size_t ws_size,
                              hipStream_t stream) {
    const float*         q    = (const float*)d_in[0];
    const float*         k    = (const float*)d_in[1];
    const float*         v    = (const float*)d_in[2];
    const unsigned char* mask = (const unsigned char*)d_in[3];  // jnp bool_ = 1 byte
    float* out = (float*)d_out;

    (void)in_sizes; (void)n_in; (void)out_size; (void)d_ws; (void)ws_size;

    // >64KB dynamic LDS: opt in (idempotent, not a stream op -> graph-capture safe)
    hipFuncSetAttribute((const void*)general_attention_kernel,
                        hipFuncAttributeMaxDynamicSharedMemorySize, (int)LDS_BYTES);

    dim3 grid(B_ * (LQ / ROWS));   // 256 workgroups: one 16-query stripe each
    general_attention_kernel<<<grid, NTHREADS, LDS_BYTES, stream>>>(q, k, v, mask, out);
}